// MambaSplit_14010183319590
// MI455X (gfx1250) — compile-verified
//
#include <hip/hip_runtime.h>
#include <hip/hip_bf16.h>

// MambaSplit / SS2D forward for MI455X (gfx1250, wave32).
// Matmuls via V_WMMA_F32_16X16X4_F32; weights staged into LDS with
// GLOBAL_LOAD_ASYNC_TO_LDS_B128 (ASYNCcnt) to decouple WMMA issue from VMEM latency.

typedef float v2f_ __attribute__((ext_vector_type(2)));
typedef float v8f_ __attribute__((ext_vector_type(8)));

#define B_   4
#define C_   128
#define L_   4096
#define D_   64
#define DI_  128
#define N_   16
#define K_   4

__device__ __forceinline__ float sigm_(float x) { return 1.0f / (1.0f + __expf(-x)); }
__device__ __forceinline__ float softplus_(float x) { return (x > 20.0f) ? x : log1pf(__expf(x)); }

// -------------------- depthwise 3x3 conv (pad 1) --------------------
__global__ void dwconv_kernel(const float* __restrict__ in, const float* __restrict__ wgt,
                              const float* __restrict__ bias, float* __restrict__ out,
                              float* __restrict__ out_t, int nch, int act_silu)
{
  const int bc = blockIdx.x;
  const int c  = bc % nch;
  const int b  = bc / nch;
  const float* src = in + (size_t)bc * L_;
  float k[9];
#pragma unroll
  for (int i = 0; i < 9; ++i) k[i] = wgt[c * 9 + i];
  const float bv = bias[c];
  for (int p = threadIdx.x; p < L_; p += blockDim.x) {
    const int h = p >> 6, w = p & 63;
    float acc = bv;
#pragma unroll
    for (int dy = -1; dy <= 1; ++dy) {
      const int hy = h + dy;
      if (hy < 0 || hy > 63) continue;
#pragma unroll
      for (int dx = -1; dx <= 1; ++dx) {
        const int wx = w + dx;
        if (wx < 0 || wx > 63) continue;
        acc += k[(dy + 1) * 3 + (dx + 1)] * src[hy * 64 + wx];
      }
    }
    if (act_silu) acc = acc * sigm_(acc);
    if (out)   out[(size_t)bc * L_ + p] = acc;
    if (out_t) out_t[((size_t)b * L_ + p) * nch + c] = acc;
  }
}

// -------------------- weight prep --------------------
__global__ void prep_kernel(const float* __restrict__ dt_w, const float* __restrict__ x_proj_w,
                            const float* __restrict__ A_logs, float* __restrict__ Wcat,
                            float* __restrict__ A_neg)
{
  const int i = blockIdx.x * blockDim.x + threadIdx.x;
  if (i < 640 * 128) {
    const int row = i / 128, dp = i % 128;
    const int k = row / 160, c = row % 160;
    float v;
    if (c < 128) {
      v = 0.0f;
#pragma unroll
      for (int r = 0; r < 4; ++r)
        v += dt_w[(k * 128 + c) * 4 + r] * x_proj_w[(k * 36 + r) * 128 + dp];
    } else {
      v = x_proj_w[(k * 36 + 4 + (c - 128)) * 128 + dp];
    }
    Wcat[i] = v;
  }
  if (i < K_ * DI_ * N_) A_neg[i] = -__expf(A_logs[i]);
}

// -------------------- LayerNorm over 64 channels of one branch --------------------
__global__ void ln1_kernel(const float* __restrict__ xd, const float* __restrict__ g,
                           const float* __restrict__ bb, float* __restrict__ xn, int c0)
{
  const int idx = blockIdx.x * blockDim.x + threadIdx.x;   // b*L + p
  const int b = idx >> 12, p = idx & 4095;
  const float* src = xd + ((size_t)(b * C_ + c0)) * L_ + p;
  float s = 0.0f, s2 = 0.0f;
  for (int c = 0; c < 64; ++c) { const float v = src[(size_t)c * L_]; s += v; s2 += v * v; }
  const float mu = s * (1.0f / 64.0f);
  const float var = s2 * (1.0f / 64.0f) - mu * mu;
  const float rs = rsqrtf(var + 1e-5f);
  float* dst = xn + (size_t)idx * 64;
  for (int c = 0; c < 64; ++c)
    dst[c] = (src[(size_t)c * L_] - mu) * rs * g[c] + bb[c];
}

// -------------------- WMMA f32 16x16x4 helpers --------------------
// A 16x4: lane<16 -> row m0+lane, (K=kk,kk+1); lane>=16 -> row m0+lane-16, (K=kk+2,kk+3)
__device__ __forceinline__ v2f_ ldfrag_(const float* __restrict__ A, int lda, int m0, int kk, int lane)
{
  const int half = lane >> 4;
  const int r = m0 + (lane & 15);
  const int kb = kk + half * 2;
  v2f_ v;
  v.x = A[(size_t)r * lda + kb];
  v.y = A[(size_t)r * lda + kb + 1];
  return v;
}

// Async-stage a 64-row weight strip [n0, n0+64) x KDIM into LDS with padded row
// stride (KDIM+4 floats): 16B-aligned rows, 4-bank row stagger (conflict-free
// fragment reads). Uses GLOBAL_LOAD_ASYNC_TO_LDS_B128 + s_wait_asynccnt.
template <int KDIM>
__device__ __forceinline__ void stage_weights_async_(const float* __restrict__ Wt, int n0, float* wsh)
{
  constexpr int CH = KDIM / 4;                 // 16-byte chunks per row
  for (int t = threadIdx.x; t < 64 * CH; t += 256) {
    const int row = t / CH, ch = t % CH;
    const unsigned loff = (unsigned)(uintptr_t)(&wsh[row * (KDIM + 4) + ch * 4]);
    const unsigned long long g =
        (unsigned long long)(uintptr_t)(Wt + (size_t)(n0 + row) * KDIM + ch * 4);
    asm volatile("global_load_async_to_lds_b128 %0, %1, off"
                 :: "v"(loff), "v"(g) : "memory");
  }
  asm volatile("s_wait_asynccnt 0x0" ::: "memory");
  __syncthreads();
}

// One wave computes a 16(M) x 64(N) strip; B fragments from LDS, A from global
// with one-step double buffering so global loads overlap the 4 WMMAs.
template <int KDIM>
__device__ __forceinline__ void wmma_strip_lds_(const float* __restrict__ A, int lda,
                                                const float* wsh, int m0, int lane, v8f_ acc[4])
{
  const int half = lane >> 4;
  const int nl = lane & 15;
  v2f_ a = ldfrag_(A, lda, m0, 0, lane);
#pragma unroll
  for (int kk = 0; kk < KDIM; kk += 4) {
    v2f_ a_nxt = a;
    if (kk + 4 < KDIM) a_nxt = ldfrag_(A, lda, m0, kk + 4, lane);
    const int kb = kk + half * 2;
#pragma unroll
    for (int j = 0; j < 4; ++j) {
      const float* wp = &wsh[(16 * j + nl) * (KDIM + 4) + kb];
      v2f_ bf; bf.x = wp[0]; bf.y = wp[1];
      acc[j] = __builtin_amdgcn_wmma_f32_16x16x4_f32(false, a, false, bf, (short)0, acc[j], false, false);
    }
    a = a_nxt;
  }
}

// -------------------- GEMM1: in_proj  [16384 x 64] * [256 x 64]^T --------------------
__global__ void gemm_inproj_kernel(const float* __restrict__ xn, const float* __restrict__ W,
                                   float* __restrict__ xc, float* __restrict__ z)
{
  __shared__ float wsh[64 * (64 + 4)];
  const int lane = threadIdx.x & 31, wv = threadIdx.x >> 5;
  const int m0 = blockIdx.x * 128 + wv * 16;
  const int n0 = blockIdx.y * 64;
  stage_weights_async_<64>(W, n0, wsh);
  v8f_ acc[4] = {};
  wmma_strip_lds_<64>(xn, 64, wsh, m0, lane, acc);
  const int half = lane >> 4, nl = lane & 15;
#pragma unroll
  for (int j = 0; j < 4; ++j) {
    const int n = n0 + 16 * j + nl;
#pragma unroll
    for (int i = 0; i < 8; ++i) {
      const int row = m0 + half * 8 + i;
      const int b = row >> 12, p = row & 4095;
      const float v = acc[j][i];
      if (n < 128) xc[((size_t)(b * 128 + n)) * L_ + p] = v;
      else         z[(size_t)row * 128 + (n - 128)] = v;
    }
  }
}

// -------------------- GEMM2: fused delta/B/C  [16384 x 128] * [640 x 128]^T ----------
__global__ void gemm_xdbl_kernel(const float* __restrict__ xct, const float* __restrict__ Wcat,
                                 float* __restrict__ g)
{
  __shared__ float wsh[64 * (128 + 4)];
  const int lane = threadIdx.x & 31, wv = threadIdx.x >> 5;
  const int m0 = blockIdx.x * 128 + wv * 16;
  const int n0 = blockIdx.y * 64;
  stage_weights_async_<128>(Wcat, n0, wsh);
  v8f_ acc[4] = {};
  wmma_strip_lds_<128>(xct, 128, wsh, m0, lane, acc);
  const int half = lane >> 4, nl = lane & 15;
#pragma unroll
  for (int j = 0; j < 4; ++j) {
    const int n = n0 + 16 * j + nl;
#pragma unroll
    for (int i = 0; i < 8; ++i) {
      const int row = m0 + half * 8 + i;
      g[(size_t)row * 640 + n] = acc[j][i];
    }
  }
}

// -------------------- GEMM3: out_proj [16384 x 128] * [64 x 128]^T + residual --------
__global__ void gemm_outproj_kernel(const float* __restrict__ ya, const float* __restrict__ W,
                                    const float* __restrict__ xd, const float* __restrict__ scale,
                                    float* __restrict__ out, int c0)
{
  __shared__ float wsh[64 * (128 + 4)];
  const int lane = threadIdx.x & 31, wv = threadIdx.x >> 5;
  const int m0 = blockIdx.x * 128 + wv * 16;
  stage_weights_async_<128>(W, 0, wsh);
  v8f_ acc[4] = {};
  wmma_strip_lds_<128>(ya, 128, wsh, m0, lane, acc);
  const float sc1 = 1.0f + scale[0];
  const int half = lane >> 4, nl = lane & 15;
#pragma unroll
  for (int j = 0; j < 4; ++j) {
    const int n = 16 * j + nl;
#pragma unroll
    for (int i = 0; i < 8; ++i) {
      const int row = m0 + half * 8 + i;
      const int b = row >> 12, p = row & 4095;
      const size_t oidx = ((size_t)(b * C_ + c0 + n)) * L_ + p;
      out[oidx] = sc1 * xd[oidx] + acc[j][i];
    }
  }
}

// -------------------- selective scan --------------------
__device__ __forceinline__ int scan_pos_(int k, int l)
{
  const int t = (k & 2) ? (L_ - 1 - l) : l;
  return (k & 1) ? (((t & 63) << 6) | (t >> 6)) : t;
}

__global__ void scan_kernel(const float* __restrict__ g, const float* __restrict__ xct,
                            const float* __restrict__ dt_b, const float* __restrict__ A_neg,
                            const float* __restrict__ Ds, float* __restrict__ out_y)
{
  const int bk = blockIdx.x;
  const int b = bk >> 2, k = bk & 3;
  const int d = threadIdx.x;
  float A[N_];
#pragma unroll
  for (int n = 0; n < N_; ++n) A[n] = A_neg[(k * DI_ + d) * N_ + n];
  const float dtb = dt_b[k * DI_ + d];
  const float Dp  = Ds[k * DI_ + d];
  float h[N_];
#pragma unroll
  for (int n = 0; n < N_; ++n) h[n] = 0.0f;

  for (int l = 0; l < L_; ++l) {
    const int p = scan_pos_(k, l);
    const float* __restrict__ grow = g + (size_t)(b * L_ + p) * 640 + k * 160;
    if (l + 1 < L_) {
      const int pn = scan_pos_(k, l + 1);
      __builtin_prefetch(g + (size_t)(b * L_ + pn) * 640 + k * 160, 0, 0);
      __builtin_prefetch(xct + ((size_t)(b * L_ + pn)) * DI_, 0, 0);
    }
    const float db = softplus_(grow[d] + dtb);
    const float u  = xct[((size_t)(b * L_ + p)) * DI_ + d];
    const float du = db * u;
    float y = 0.0f;
#pragma unroll
    for (int n = 0; n < N_; ++n) {
      const float Bv = grow[128 + n];      // thread-uniform -> scalar loads
      const float Cv = grow[144 + n];
      h[n] = __expf(db * A[n]) * h[n] + du * Bv;
      y += h[n] * Cv;
    }
    out_y[(((size_t)k * B_ + b) * L_ + p) * DI_ + d] = y + Dp * u;
  }
}

// -------------------- combine 4 dirs + out LayerNorm(128) + SiLU(z) gate --------------------
__global__ void combine_kernel(const float* __restrict__ out_y, const float* __restrict__ z,
                               const float* __restrict__ ong, const float* __restrict__ onb,
                               float* __restrict__ ya)
{
  const int bp = blockIdx.x;            // b*L + p
  const int b = bp >> 12, p = bp & 4095;
  const int d = threadIdx.x;
  float v = 0.0f;
#pragma unroll
  for (int k = 0; k < K_; ++k)
    v += out_y[(((size_t)k * B_ + b) * L_ + p) * DI_ + d];

  __shared__ float red[DI_];
  red[d] = v; __syncthreads();
  for (int s = 64; s > 0; s >>= 1) { if (d < s) red[d] += red[d + s]; __syncthreads(); }
  const float mu = red[0] * (1.0f / 128.0f);
  __syncthreads();
  red[d] = v * v; __syncthreads();
  for (int s = 64; s > 0; s >>= 1) { if (d < s) red[d] += red[d + s]; __syncthreads(); }
  const float var = red[0] * (1.0f / 128.0f) - mu * mu;
  const float rs = rsqrtf(var + 1e-5f);

  const float w = (v - mu) * rs * ong[d] + onb[d];
  const float zd = z[(size_t)bp * DI_ + d];
  ya[(size_t)bp * DI_ + d] = w * (zd * sigm_(zd));
}

// -------------------- instance-norm stats over (H,W) per (b,c) --------------------
__global__ void instats_kernel(const float* __restrict__ xo, float* __restrict__ stats)
{
  const int bc = blockIdx.x;
  float s = 0.0f, s2 = 0.0f;
  for (int p = threadIdx.x; p < L_; p += blockDim.x) {
    const float v = xo[(size_t)bc * L_ + p];
    s += v; s2 += v * v;
  }
  __shared__ float r1[256], r2[256];
  r1[threadIdx.x] = s; r2[threadIdx.x] = s2; __syncthreads();
  for (int st = 128; st > 0; st >>= 1) {
    if (threadIdx.x < st) { r1[threadIdx.x] += r1[threadIdx.x + st]; r2[threadIdx.x] += r2[threadIdx.x + st]; }
    __syncthreads();
  }
  if (threadIdx.x == 0) {
    const float mu = r1[0] * (1.0f / (float)L_);
    const float var = r2[0] * (1.0f / (float)L_) - mu * mu;
    stats[2 * bc]     = mu;
    stats[2 * bc + 1] = rsqrtf(var + 1e-5f);
  }
}

// -------------------- final normalize + leaky relu (in place on d_out) --------------------
__global__ void finalnorm_kernel(float* __restrict__ xo, const float* __restrict__ stats,
                                 const float* __restrict__ gma, const float* __restrict__ bta)
{
  const int i = blockIdx.x * blockDim.x + threadIdx.x;
  const int bc = i >> 12;
  const int c = bc & 127;
  const float v = (xo[i] - stats[2 * bc]) * stats[2 * bc + 1] * gma[c] + bta[c];
  xo[i] = (v > 0.0f) ? v : 0.01f * v;
}

// -------------------- host launch --------------------
extern "C" void kernel_launch(void* const* d_in, const int* in_sizes, int n_in,
                              void* d_out, int out_size, void* d_ws, size_t ws_size,
                              hipStream_t stream)
{
  (void)in_sizes; (void)n_in; (void)out_size; (void)ws_size;
  const float* x          = (const float*)d_in[0];
  const float* dw_w       = (const float*)d_in[1];
  const float* dw_b       = (const float*)d_in[2];
  const float* scale      = (const float*)d_in[3];
  const float* ln1_g      = (const float*)d_in[4];
  const float* ln1_b      = (const float*)d_in[5];
  const float* in_proj_w  = (const float*)d_in[6];
  const float* conv_w     = (const float*)d_in[7];
  const float* conv_b     = (const float*)d_in[8];
  const float* x_proj_w   = (const float*)d_in[9];
  const float* dt_w       = (const float*)d_in[10];
  const float* dt_b       = (const float*)d_in[11];
  const float* A_logs     = (const float*)d_in[12];
  const float* Ds         = (const float*)d_in[13];
  const float* on_g       = (const float*)d_in[14];
  const float* on_b       = (const float*)d_in[15];
  const float* out_proj_w = (const float*)d_in[16];
  const float* in_gamma   = (const float*)d_in[17];
  const float* in_beta    = (const float*)d_in[18];
  float* out = (float*)d_out;

  float* wsf   = (float*)d_ws;
  float* xd    = wsf;                    // 2,097,152
  float* Aneg  = xd + 2097152;           // 8,192
  float* Wcat  = Aneg + 8192;            // 81,920
  float* stats = Wcat + 81920;           // 1,024
  float* xn    = stats + 1024;           // 1,048,576
  float* xc    = xn + 1048576;           // 2,097,152
  float* zbuf  = xc + 2097152;           // 2,097,152
  float* xct   = zbuf + 2097152;         // 2,097,152
  float* gbuf  = xct + 2097152;          // 10,485,760
  float* outy  = gbuf + 10485760;        // 8,388,608
  float* ya    = outy + 8388608;         // 2,097,152  (total ~122 MB)

  dwconv_kernel<<<B_ * C_, 256, 0, stream>>>(x, dw_w, dw_b, xd, nullptr, C_, 0);
  prep_kernel<<<(640 * 128 + 255) / 256, 256, 0, stream>>>(dt_w, x_proj_w, A_logs, Wcat, Aneg);

  for (int br = 0; br < 2; ++br) {
    const int c0 = br * 64;
    ln1_kernel<<<(B_ * L_) / 256, 256, 0, stream>>>(xd, ln1_g, ln1_b, xn, c0);
    gemm_inproj_kernel<<<dim3(128, 4), 256, 0, stream>>>(xn, in_proj_w, xc, zbuf);
    dwconv_kernel<<<B_ * DI_, 256, 0, stream>>>(xc, conv_w, conv_b, nullptr, xct, DI_, 1);
    gemm_xdbl_kernel<<<dim3(128, 10), 256, 0, stream>>>(xct, Wcat, gbuf);
    scan_kernel<<<B_ * K_, DI_, 0, stream>>>(gbuf, xct, dt_b, Aneg, Ds, outy);
    combine_kernel<<<B_ * L_, DI_, 0, stream>>>(outy, zbuf, on_g, on_b, ya);
    gemm_outproj_kernel<<<dim3(128, 1), 256, 0, stream>>>(ya, out_proj_w, xd, scale, out, c0);
  }

  instats_kernel<<<B_ * C_, 256, 0, stream>>>(out, stats);
  finalnorm_kernel<<<(B_ * C_ * L_) / 256, 256, 0, stream>>>(out, stats, in_gamma, in_beta);
}